// SugenoFuzzyIntegral_90941637525597
// MI455X (gfx1250) — compile-verified
//
#include <hip/hip_runtime.h>
#include <hip/hip_bf16.h>
#include <stdint.h>

#define NPAD   1024
#define BLOCK  256
#define NWAVES (BLOCK / 32)

typedef __attribute__((ext_vector_type(4))) unsigned int u32x4_t;
typedef __attribute__((ext_vector_type(8))) int          i32x8_t;
typedef __attribute__((ext_vector_type(4))) int          i32x4_t;

// register-register compare-exchange: lower index element in x, upper in y
#define CE_PAIR(x, y, d)                                   \
    {                                                      \
        float _mn = fminf((x), (y)), _mx = fmaxf((x), (y));\
        (x) = (d) ? _mx : _mn;                             \
        (y) = (d) ? _mn : _mx;                             \
    }

// cross-lane compare-exchange via shfl_xor; idx is this element's global index
__device__ __forceinline__ float ce_x(float v, int idx, int j, int k)
{
    float p = __shfl_xor(v, j, 32);
    bool keepmax = (((idx & j) == 0) == ((idx & k) == 0));
    return keepmax ? fmaxf(v, p) : fminf(v, p);
}

__global__ __launch_bounds__(BLOCK)
void SugenoFuzzyIntegral_kernel(const float* __restrict__ mu,
                                const int*   __restrict__ target_class,
                                const float* __restrict__ log_lambda,
                                float*       __restrict__ out,
                                int C)
{
    __shared__ float smem[NPAD];
    __shared__ float wAs[NWAVES];
    __shared__ float wBs[NWAVES];
    __shared__ float sh_tmu;

    const int tid  = threadIdx.x;
    const int row  = blockIdx.x;
    const int lane = tid & 31;
    const int wid  = tid >> 5;

    // ---- TDM: DMA this row (C floats) from global into LDS (wave 0 issues) ----
    if (tid < 32) {
        uint64_t gaddr = (uint64_t)(uintptr_t)(mu + (size_t)row * (size_t)C);
        uint32_t laddr = (uint32_t)(uintptr_t)&smem[0];

        u32x4_t g0 = { 1u,                                   // count=1
                       laddr,                                // lds_addr
                       (uint32_t)gaddr,                      // global_addr lo
                       (((uint32_t)(gaddr >> 32)) & 0x01ffffffu) | (2u << 30) }; // hi | type=2

        i32x8_t g1 = { (int)(2u << 16),                                     // data_size = 4B
                       (int)(((uint32_t)C & 0xffffu) << 16),                // tensor_dim0 lo
                       (int)((1u << 16) | (((uint32_t)C >> 16) & 0xffffu)), // tensor_dim1=1 | dim0 hi
                       (int)(((uint32_t)C & 0xffffu) << 16),                // tile_dim0 = C
                       1,                                                   // tile_dim1 = 1
                       C,                                                   // tensor_dim0_stride lo
                       0, 0 };
        i32x4_t g2 = { 0, 0, 0, 0 };
        i32x4_t g3 = { 0, 0, 0, 0 };
#if defined(__clang_major__) && __clang_major__ >= 23
        i32x8_t g4 = { 0, 0, 0, 0, 0, 0, 0, 0 };
        __builtin_amdgcn_tensor_load_to_lds(g0, g1, g2, g3, g4, 0);
#else
        __builtin_amdgcn_tensor_load_to_lds(g0, g1, g2, g3, 0);
#endif
    }

    // Pad tail with -inf while the DMA is in flight (disjoint LDS region).
    for (int i = C + tid; i < NPAD; i += BLOCK) smem[i] = -__builtin_inff();

    if (tid < 32) __builtin_amdgcn_s_wait_tensorcnt(0);
    __syncthreads();

    // Grab target_mu before the sort destroys index association.
    if (tid == 0) sh_tmu = smem[target_class[row]];
    __syncthreads();

    // ---- Bitonic sort (descending). Each wave owns 128 elements in registers. ----
    const int base = wid * 128;
    const int i0 = base + lane;
    const int i1 = i0 + 32;
    const int i2 = i0 + 64;
    const int i3 = i0 + 96;

    float a0 = smem[i0], a1 = smem[i1], b0 = smem[i2], b1 = smem[i3];

    // stages k = 2..32 : lane-local (shfl) only
    #pragma unroll
    for (int k = 2; k <= 32; k <<= 1) {
        #pragma unroll
        for (int j = k >> 1; j >= 1; j >>= 1) {
            a0 = ce_x(a0, i0, j, k);
            a1 = ce_x(a1, i1, j, k);
            b0 = ce_x(b0, i2, j, k);
            b1 = ce_x(b1, i3, j, k);
        }
    }

    // stage k = 64 : j=32 is a register pair within each 64-block, then shfl tail
    {
        const bool dA = ((i0 & 64) == 0);
        const bool dB = ((i2 & 64) == 0);
        CE_PAIR(a0, a1, dA);
        CE_PAIR(b0, b1, dB);
        #pragma unroll
        for (int j = 16; j >= 1; j >>= 1) {
            a0 = ce_x(a0, i0, j, 64);
            a1 = ce_x(a1, i1, j, 64);
            b0 = ce_x(b0, i2, j, 64);
            b1 = ce_x(b1, i3, j, 64);
        }
    }

    // merge of the wave's 128 elements for stage k (j = 64..1), k >= 128
    auto merge128 = [&](int k) {
        const bool d = ((base & k) == 0);       // uniform: offsets < 128 <= k
        CE_PAIR(a0, b0, d); CE_PAIR(a1, b1, d); // j = 64
        CE_PAIR(a0, a1, d); CE_PAIR(b0, b1, d); // j = 32
        #pragma unroll
        for (int j = 16; j >= 1; j >>= 1) {
            a0 = ce_x(a0, i0, j, k);
            a1 = ce_x(a1, i1, j, k);
            b0 = ce_x(b0, i2, j, k);
            b1 = ce_x(b1, i3, j, k);
        }
    };

    auto store_regs = [&]() {
        smem[i0] = a0; smem[i1] = a1; smem[i2] = b0; smem[i3] = b1;
    };
    auto load_regs = [&]() {
        a0 = smem[i0]; a1 = smem[i1]; b0 = smem[i2]; b1 = smem[i3];
    };

    // cross-wave LDS phase (j >= 128), owner-updates-pair mapping
    auto lds_phase = [&](int k, int j) {
        #pragma unroll
        for (int u = 0; u < NPAD / (2 * BLOCK); ++u) {
            int t = tid + u * BLOCK;
            int i = ((t & ~(j - 1)) << 1) | (t & (j - 1));
            int p = i | j;
            float x = smem[i];
            float y = smem[p];
            bool desc = ((i & k) == 0);
            if ((x < y) == desc) { smem[i] = y; smem[p] = x; }
        }
        __syncthreads();
    };

    merge128(128);                 // stage k = 128 fully in registers
    store_regs(); __syncthreads();

    // k = 256
    lds_phase(256, 128);
    load_regs(); merge128(256); store_regs(); __syncthreads();

    // k = 512
    lds_phase(512, 256);
    lds_phase(512, 128);
    load_regs(); merge128(512); store_regs(); __syncthreads();

    // k = 1024 (final, fully descending)
    lds_phase(1024, 512);
    lds_phase(1024, 256);
    lds_phase(1024, 128);
    load_regs(); merge128(1024); store_regs(); __syncthreads();

    // ---- Affine-scan of g_i = (1+lam*s_i)*g_{i-1} + s_i ----
    const float lam = tanhf(log_lambda[0]) * 9.99f;

    float s[4];
    #pragma unroll
    for (int m = 0; m < 4; ++m) {
        float v = smem[tid * 4 + m];
        s[m] = fminf(fmaxf(v, 0.0f), 1.0f);   // pads clip to 0 -> identity affine map
    }

    float A = 1.0f, Bv = 0.0f;
    #pragma unroll
    for (int m = 0; m < 4; ++m) {
        float a = fmaf(lam, s[m], 1.0f);
        Bv = fmaf(Bv, a, s[m]);
        A  = A * a;
    }

    #pragma unroll
    for (int off = 1; off < 32; off <<= 1) {
        float Au = __shfl_up(A,  off, 32);
        float Bu = __shfl_up(Bv, off, 32);
        if (lane >= off) {
            Bv = fmaf(Bu, A, Bv);
            A  = Au * A;
        }
    }

    if (lane == 31) { wAs[wid] = A; wBs[wid] = Bv; }
    __syncthreads();

    float Aex = __shfl_up(A,  1, 32);
    float Bex = __shfl_up(Bv, 1, 32);
    if (lane == 0) { Aex = 1.0f; Bex = 0.0f; }

    float Bc = 0.0f;
    for (int w = 0; w < wid; ++w)
        Bc = fmaf(Bc, wAs[w], wBs[w]);

    float g  = fmaf(Aex, Bc, Bex);
    float mx = 0.0f;
    #pragma unroll
    for (int m = 0; m < 4; ++m) {
        g  = fmaf(lam * g, s[m], g + s[m]);   // g + s + lam*g*s
        mx = fmaxf(mx, fminf(s[m], g));
    }

    #pragma unroll
    for (int off = 16; off >= 1; off >>= 1)
        mx = fmaxf(mx, __shfl_xor(mx, off, 32));

    __syncthreads();               // wAs reuse
    if (lane == 0) wAs[wid] = mx;
    __syncthreads();

    if (tid == 0) {
        float integral = wAs[0];
        #pragma unroll
        for (int w = 1; w < NWAVES; ++w) integral = fmaxf(integral, wAs[w]);
        float rowmax = smem[0];    // descending sort -> max(mu)
        out[row] = integral * (sh_tmu / (rowmax + 1e-8f));
    }
}

extern "C" void kernel_launch(void* const* d_in, const int* in_sizes, int n_in,
                              void* d_out, int out_size, void* d_ws, size_t ws_size,
                              hipStream_t stream)
{
    const float* mu  = (const float*)d_in[0];
    const int*   tc  = (const int*)d_in[1];
    const float* ll  = (const float*)d_in[2];
    float*       out = (float*)d_out;

    const int B = in_sizes[1];
    const int C = in_sizes[0] / (B > 0 ? B : 1);   // 1000 here; kernel supports C <= 1024

    hipLaunchKernelGGL(SugenoFuzzyIntegral_kernel, dim3(B), dim3(BLOCK), 0, stream,
                       mu, tc, ll, out, C);
}